// ObservationPointToSubBasinAggregation_46033459478826
// MI455X (gfx1250) — compile-verified
//
#include <hip/hip_runtime.h>

typedef __attribute__((ext_vector_type(2))) float v2f;
typedef __attribute__((ext_vector_type(8))) float v8f;

#define C_IN   64
#define HEADS  2
#define COLS   128   // HEADS * C_IN
#define NEG_SLOPE 0.2f

// ---- monotone float <-> uint encoding for atomic segment-max --------------
__device__ __forceinline__ unsigned enc_f32(float f) {
    unsigned b = __float_as_uint(f);
    return (b & 0x80000000u) ? ~b : (b | 0x80000000u);
}
__device__ __forceinline__ float dec_f32(unsigned u) {
    return (u & 0x80000000u) ? __uint_as_float(u & 0x7FFFFFFFu)
                             : __uint_as_float(~u);
}
__device__ __forceinline__ float leaky(float x) {
    return x > 0.0f ? x : NEG_SLOPE * x;
}

// ---- kernel 0: init out = bias, m_enc = 0, s = 0 --------------------------
__global__ __launch_bounds__(256) void gat_init(float* __restrict__ out,
                                                const float* __restrict__ bias,
                                                unsigned* __restrict__ m_enc,
                                                float* __restrict__ s,
                                                int N) {
    int idx = blockIdx.x * 256 + threadIdx.x;
    if (idx < N * COLS) out[idx] = bias[idx & (COLS - 1)];
    if (idx < N * HEADS) { m_enc[idx] = 0u; s[idx] = 0.0f; }
}

// ---- kernel 1: h = x @ W via V_WMMA_F32_16X16X4_F32 -----------------------
// one wave per 16x16 output tile; 8 waves/block cover all 128 columns
__global__ __launch_bounds__(256) void gat_gemm_wmma(const float* __restrict__ x,
                                                     const float* __restrict__ W,
                                                     float* __restrict__ h,
                                                     int N) {
    const int lane  = threadIdx.x & 31;
    const int wave  = threadIdx.x >> 5;          // 0..7 -> column tile
    const int m0    = blockIdx.x * 16;
    const int n0    = wave * 16;
    const int lhalf = lane & 15;
    const int khi   = (lane >> 4) << 1;          // 0 for lanes 0-15, 2 for 16-31

    v8f acc = {};
    const int arow = min(m0 + lhalf, N - 1);     // clamp (keeps EXEC uniform)
    #pragma unroll
    for (int kb = 0; kb < C_IN; kb += 4) {
        v2f a, b;
        // A 16x4: V0 = K(khi), V1 = K(khi+1), row M = lane%16
        a.x = x[arow * C_IN + kb + khi + 0];
        a.y = x[arow * C_IN + kb + khi + 1];
        // B 4x16: V0 = row K(khi), V1 = row K(khi+1), col N = lane%16
        b.x = W[(kb + khi + 0) * COLS + n0 + lhalf];
        b.y = W[(kb + khi + 1) * COLS + n0 + lhalf];
        acc = __builtin_amdgcn_wmma_f32_16x16x4_f32(
                  false, a, false, b, (short)0, acc, false, false);
    }
    // C/D: VGPR j holds row m0+j (lanes 0-15) / m0+8+j (lanes 16-31), col n0+lane%16
    const int rbase = m0 + ((lane >> 4) << 3);
    #pragma unroll
    for (int j = 0; j < 8; ++j) {
        int row = rbase + j;
        if (row < N) h[row * COLS + n0 + lhalf] = acc[j];
    }
}

// ---- kernel 2: per-(node,head) attention dots -----------------------------
__global__ __launch_bounds__(256) void gat_att(const float* __restrict__ h,
                                               const float* __restrict__ att_src,
                                               const float* __restrict__ att_dst,
                                               float* __restrict__ a_s,
                                               float* __restrict__ a_d,
                                               int N) {
    int idx = blockIdx.x * 256 + threadIdx.x;    // node*HEADS + head
    if (idx >= N * HEADS) return;
    int node = idx >> 1, head = idx & 1;
    const float* hp = h + node * COLS + head * C_IN;
    const float* as = att_src + head * C_IN;
    const float* ad = att_dst + head * C_IN;
    float vs = 0.0f, vd = 0.0f;
    #pragma unroll 8
    for (int c = 0; c < C_IN; ++c) {
        float v = hp[c];
        vs += v * as[c];
        vd += v * ad[c];
    }
    a_s[idx] = vs;
    a_d[idx] = vd;
}

// ---- kernel 3: segment max of leaky-relu logits (atomic u32 max) ----------
__global__ __launch_bounds__(256) void gat_max(const int* __restrict__ ei,
                                               const float* __restrict__ a_s,
                                               const float* __restrict__ a_d,
                                               unsigned* __restrict__ m_enc,
                                               int E, int N) {
    int idx = blockIdx.x * 256 + threadIdx.x;    // (edge)*HEADS + head
    int Etot = E + N;
    if (idx >= Etot * HEADS) return;
    int e = idx >> 1, hd = idx & 1;
    int srcn, dstn;
    if (e < E) { srcn = ei[e]; dstn = ei[E + e]; }
    else       { srcn = dstn = e - E; }          // self loop
    float lg = leaky(a_s[srcn * HEADS + hd] + a_d[dstn * HEADS + hd]);
    atomicMax(&m_enc[dstn * HEADS + hd], enc_f32(lg));
}

// ---- kernel 4: segment sum of exp(logit - max) ----------------------------
__global__ __launch_bounds__(256) void gat_expsum(const int* __restrict__ ei,
                                                  const float* __restrict__ a_s,
                                                  const float* __restrict__ a_d,
                                                  const unsigned* __restrict__ m_enc,
                                                  float* __restrict__ s,
                                                  int E, int N) {
    int idx = blockIdx.x * 256 + threadIdx.x;
    int Etot = E + N;
    if (idx >= Etot * HEADS) return;
    int e = idx >> 1, hd = idx & 1;
    int srcn, dstn;
    if (e < E) { srcn = ei[e]; dstn = ei[E + e]; }
    else       { srcn = dstn = e - E; }
    float lg = leaky(a_s[srcn * HEADS + hd] + a_d[dstn * HEADS + hd]);
    float ev = __expf(lg - dec_f32(m_enc[dstn * HEADS + hd]));
    atomicAdd(&s[dstn * HEADS + hd], ev);
}

// ---- kernel 5: weighted message scatter (warp per edge, float4 per lane) --
__global__ __launch_bounds__(256) void gat_scatter(const int* __restrict__ ei,
                                                   const float* __restrict__ h,
                                                   const float* __restrict__ a_s,
                                                   const float* __restrict__ a_d,
                                                   const unsigned* __restrict__ m_enc,
                                                   const float* __restrict__ s,
                                                   float* __restrict__ out,
                                                   int E, int N) {
    int e = blockIdx.x * 8 + (threadIdx.x >> 5);
    int Etot = E + N;
    if (e >= Etot) return;
    int lane = threadIdx.x & 31;
    int srcn, dstn;
    if (e < E) { srcn = ei[e]; dstn = ei[E + e]; }
    else       { srcn = dstn = e - E; }
    int hd = lane >> 4;                           // lanes 0-15: head0, 16-31: head1
    float lg = leaky(a_s[srcn * HEADS + hd] + a_d[dstn * HEADS + hd]);
    float alpha = __expf(lg - dec_f32(m_enc[dstn * HEADS + hd]))
                / (s[dstn * HEADS + hd] + 1e-16f);
    const float4 hv = *(const float4*)(h + srcn * COLS + lane * 4);
    float* op = out + dstn * COLS + lane * 4;
    atomicAdd(op + 0, hv.x * alpha);
    atomicAdd(op + 1, hv.y * alpha);
    atomicAdd(op + 2, hv.z * alpha);
    atomicAdd(op + 3, hv.w * alpha);
}

// ---------------------------------------------------------------------------
extern "C" void kernel_launch(void* const* d_in, const int* in_sizes, int n_in,
                              void* d_out, int out_size, void* d_ws, size_t ws_size,
                              hipStream_t stream) {
    const float* x       = (const float*)d_in[0];
    const int*   ei      = (const int*)  d_in[1];   // [2, E] row-major: src row then dst row
    const float* W       = (const float*)d_in[2];
    const float* att_src = (const float*)d_in[3];
    const float* att_dst = (const float*)d_in[4];
    const float* bias    = (const float*)d_in[5];
    float*       out     = (float*)d_out;

    const int N = in_sizes[0] / C_IN;   // 50000
    const int E = in_sizes[1] / 2;      // 800000
    const int Etot = E + N;

    // workspace carve-up (aligned to 256B)
    char* ws = (char*)d_ws;
    size_t off = 0;
    auto carve = [&](size_t bytes) {
        void* p = ws + off;
        off = (off + bytes + 255) & ~size_t(255);
        return p;
    };
    float*    h     = (float*)   carve((size_t)N * COLS  * sizeof(float));
    float*    a_s   = (float*)   carve((size_t)N * HEADS * sizeof(float));
    float*    a_d   = (float*)   carve((size_t)N * HEADS * sizeof(float));
    unsigned* m_enc = (unsigned*)carve((size_t)N * HEADS * sizeof(unsigned));
    float*    s     = (float*)   carve((size_t)N * HEADS * sizeof(float));
    (void)ws_size; (void)n_in; (void)out_size;

    // 0) init accumulators (out = bias broadcast; m = enc(-inf) lower bound; s = 0)
    gat_init<<<(N * COLS + 255) / 256, 256, 0, stream>>>(out, bias, m_enc, s, N);

    // 1) h = x @ W  (WMMA f32 16x16x4)
    gat_gemm_wmma<<<(N + 15) / 16, 256, 0, stream>>>(x, W, h, N);

    // 2) per-node attention dots
    gat_att<<<(N * HEADS + 255) / 256, 256, 0, stream>>>(h, att_src, att_dst, a_s, a_d, N);

    // 3) segment max
    gat_max<<<(Etot * HEADS + 255) / 256, 256, 0, stream>>>(ei, a_s, a_d, m_enc, E, N);

    // 4) segment exp-sum
    gat_expsum<<<(Etot * HEADS + 255) / 256, 256, 0, stream>>>(ei, a_s, a_d, m_enc, s, E, N);

    // 5) weighted scatter-sum (one wave per edge)
    gat_scatter<<<(Etot + 7) / 8, 256, 0, stream>>>(ei, h, a_s, a_d, m_enc, s, out, E, N);
}